// CRFLoss_ma_71631464563256
// MI455X (gfx1250) — compile-verified
//
#include <hip/hip_runtime.h>
#include <stdint.h>

// Problem constants (from the reference).
#define A_DIM      8
#define S_DIM      200
#define B_DIM      64
#define T_DIM      32
#define START_TAG  30
#define END_TAG    31

#define NBUF       12                 // ring depth of 4KB tile buffers (48KB LDS)
#define TILE_ELEMS (T_DIM * T_DIM)    // 1024 fp32
#define TILE_BYTES (TILE_ELEMS * 4)   // 4096

// cpol temporal hint: NT (non-temporal) — the 419MB scores stream is read once
// and exceeds the 192MB L2; don't let it evict resident data.
#define TDM_CPOL   1

typedef uint32_t u32x4 __attribute__((ext_vector_type(4)));
typedef int      i32x4 __attribute__((ext_vector_type(4)));
typedef int      i32x8 __attribute__((ext_vector_type(8)));

// TDM availability: only in the device pass, and only if the builtin exists.
#if defined(__HIP_DEVICE_COMPILE__) && __has_builtin(__builtin_amdgcn_tensor_load_to_lds)
#define HAVE_TDM 1
#else
#define HAVE_TDM 0
#endif

#if HAVE_TDM
#if __has_builtin(__builtin_amdgcn_s_wait_tensorcnt)
#define WAIT_TCNT(n) __builtin_amdgcn_s_wait_tensorcnt(n)
#else
#define WAIT_TCNT(n) asm volatile("s_wait_tensorcnt %0" ::"i"(n))
#endif
#endif

__device__ __forceinline__ uint32_t lds_off_u32(const void* p) {
  // Generic pointer to LDS: flat addr low 32 bits == LDS byte offset (aperture rule).
  return (uint32_t)(uintptr_t)p;
}

// Issue one TDM DMA: 1024 contiguous fp32 (one 32x32 transition tile) -> LDS.
// D# per CDNA5 ISA ch.8: group0 = {flags/count, lds_addr, global_addr, type=2},
// group1 = {data_size=4B, tensor_dim0=1024, tensor_dim1=1, tile_dim0=1024,
//           tile_dim1=1, tensor_dim0_stride=1024}. Groups 2/3 unused (2D).
__device__ __forceinline__ void tdm_load_tile(const float* gsrc, uint32_t lds_off) {
#if HAVE_TDM
  uint64_t ga = (uint64_t)(uintptr_t)gsrc;
  u32x4 g0 = { 1u,                                    // count=1 (valid user descriptor)
               lds_off,                               // lds_addr (bytes)
               (uint32_t)ga,                          // global_addr[31:0]
               (uint32_t)(((ga >> 32) & 0x01FFFFFFu)  // global_addr[56:32]
                          | 0x80000000u) };           // type=2 ("image")
  i32x8 g1 = { (int)(2u << 16),                       // data_size=2 -> 4 bytes
               (int)(((uint32_t)TILE_ELEMS & 0xFFFFu) << 16),           // tensor_dim0 lo
               (int)((((uint32_t)TILE_ELEMS >> 16) & 0xFFFFu) | (1u << 16)), // dim0 hi | tensor_dim1 lo
               (int)(((uint32_t)TILE_ELEMS & 0xFFFFu) << 16),           // tensor_dim1 hi=0 | tile_dim0
               1,                                     // tile_dim1=1, tile_dim2=0
               TILE_ELEMS,                            // tensor_dim0_stride lo
               0, 0 };                                // stride hi / dim1_stride
  i32x4 z4 = {0, 0, 0, 0};
#if defined(__clang_major__) && (__clang_major__ >= 23)
  i32x8 z8 = {0, 0, 0, 0, 0, 0, 0, 0};
  __builtin_amdgcn_tensor_load_to_lds(g0, g1, z4, z4, z8, TDM_CPOL);
#else
  __builtin_amdgcn_tensor_load_to_lds(g0, g1, z4, z4, TDM_CPOL);
#endif
#else
  (void)gsrc; (void)lds_off;
#endif
}

// One wave32 per (a,b) chain. Lane t owns destination tag t (T==32==wave size).
__global__ __launch_bounds__(32) void crf_forward_kernel(
    const float* __restrict__ scores,   // (A,S,B,T,T)
    const int* __restrict__ targets,    // (A,S,B)
    const uint8_t* __restrict__ mask,   // (S,B) bool
    const uint8_t* __restrict__ a_mask, // (A,B) bool
    float* __restrict__ partial) {      // (A*B,)
  __shared__ __align__(16) float   tile[NBUF][TILE_ELEMS]; // 48 KB ring
  __shared__ __align__(16) float   p_lds[T_DIM];           // partition vector
  __shared__ int                   tgt_lds[S_DIM];
  __shared__ uint8_t               msk_lds[S_DIM];

  const int lane = threadIdx.x;
  const int ab   = blockIdx.x;
  const int a    = ab / B_DIM;
  const int b    = ab % B_DIM;

  // Preload per-chain targets & mask so the hot loop has no global loads.
  for (int s = lane; s < S_DIM; s += 32) {
    tgt_lds[s] = targets[(a * S_DIM + s) * B_DIM + b];
    msk_lds[s] = mask[s * B_DIM + b];
  }
  __syncthreads();

  // Tile for step s lives at base + s*B*1024 (contiguous 4KB each).
  const float* base = scores + ((size_t)a * S_DIM * B_DIM + b) * TILE_ELEMS;

#if HAVE_TDM
  // Prologue: fill the ring.
  for (int s0 = 0; s0 < NBUF; ++s0)
    tdm_load_tile(base + (size_t)s0 * B_DIM * TILE_ELEMS, lds_off_u32(&tile[s0][0]));
  WAIT_TCNT(NBUF - 1);                 // in-order completion => tile 0 ready
  const float* t0 = tile[0];
#else
  const float* t0 = base;
#endif

  // ---- step 0: partition = scores[a,0,b,START_TAG,:]; gold gather ----
  float tg = 0.0f;
  float p  = t0[START_TAG * T_DIM + lane];
  if (lane == 0 && msk_lds[0]) tg = t0[tgt_lds[0]];
  p_lds[lane] = p;
  __syncthreads();

  // ---- steps 1..S-1: log-sum-exp forward recursion ----
  for (int s = 1; s < S_DIM; ++s) {
#if HAVE_TDM
    const int pre = s - 1 + NBUF;      // refill the slot step (s-1) just freed
    if (pre < S_DIM) {
      asm volatile("s_wait_dscnt 0" ::: "memory"); // our LDS reads of that slot are done
      tdm_load_tile(base + (size_t)pre * B_DIM * TILE_ELEMS,
                    lds_off_u32(&tile[pre % NBUF][0]));
    }
    if (s + NBUF <= S_DIM) { WAIT_TCNT(NBUF - 1); }  // steady state: tile s complete
    else                   { WAIT_TCNT(0); }         // drain tail (last NBUF-1 steps)
    const float* tl = tile[s % NBUF];
#else
    const float* tl = base + (size_t)s * B_DIM * TILE_ELEMS;
#endif
    if (lane == 0 && msk_lds[s]) tg += tl[tgt_lds[s]];

    // v[f] = tile[f][lane] + partition[f]; broadcast p via uniform float4 LDS reads.
    float v[T_DIM];
#pragma unroll
    for (int f4 = 0; f4 < T_DIM / 4; ++f4) {
      const float4 pv = *(const float4*)(p_lds + 4 * f4);
      v[4 * f4 + 0] = tl[(4 * f4 + 0) * T_DIM + lane] + pv.x;
      v[4 * f4 + 1] = tl[(4 * f4 + 1) * T_DIM + lane] + pv.y;
      v[4 * f4 + 2] = tl[(4 * f4 + 2) * T_DIM + lane] + pv.z;
      v[4 * f4 + 3] = tl[(4 * f4 + 3) * T_DIM + lane] + pv.w;
    }
    float m = v[0];
#pragma unroll
    for (int f = 1; f < T_DIM; ++f) m = fmaxf(m, v[f]);
    float sum = 0.0f;
#pragma unroll
    for (int f = 0; f < T_DIM; ++f) sum += __expf(v[f] - m);
    const float np = m + __logf(sum);
    if (msk_lds[s]) p = np;            // masked update (wave-uniform branch)

    __syncthreads();                   // fence: all reads of p_lds done
    p_lds[lane] = p;
    __syncthreads();
  }

  if (lane == 0) {
    const float logZ = p_lds[END_TAG];
    partial[ab] = a_mask[a * B_DIM + b] ? (logZ - tg) : 0.0f;
  }
}

// Deterministic 512 -> 1 tree reduction; applies the /batch scale.
__global__ __launch_bounds__(256) void crf_reduce_kernel(
    const float* __restrict__ partial, float* __restrict__ out) {
  __shared__ float sm[256];
  const int t = threadIdx.x;
  sm[t] = partial[t] + partial[t + 256];
  __syncthreads();
  for (int w = 128; w > 0; w >>= 1) {
    if (t < w) sm[t] += sm[t + w];
    __syncthreads();
  }
  if (t == 0) out[0] = sm[0] * (1.0f / (float)B_DIM);
}

extern "C" void kernel_launch(void* const* d_in, const int* in_sizes, int n_in,
                              void* d_out, int out_size, void* d_ws, size_t ws_size,
                              hipStream_t stream) {
  (void)in_sizes; (void)n_in; (void)out_size; (void)ws_size;
  const float*   scores  = (const float*)d_in[0];
  const int*     targets = (const int*)d_in[1];
  const uint8_t* mask    = (const uint8_t*)d_in[2];  // numpy bool layout: 1 byte
  const uint8_t* a_mask  = (const uint8_t*)d_in[3];
  float*         partial = (float*)d_ws;             // 512 floats scratch

  crf_forward_kernel<<<A_DIM * B_DIM, 32, 0, stream>>>(scores, targets, mask,
                                                       a_mask, partial);
  crf_reduce_kernel<<<1, 256, 0, stream>>>(partial, (float*)d_out);
}